// Net_18107582120151
// MI455X (gfx1250) — compile-verified
//
#include <hip/hip_runtime.h>

typedef __attribute__((ext_vector_type(2))) float v2f;
typedef __attribute__((ext_vector_type(8))) float v8f;

#define WMMA4(a, b, c) \
    __builtin_amdgcn_wmma_f32_16x16x4_f32(false, (a), false, (b), (short)0, (c), false, false)

// ODE RHS: f([t, y0, y1]) through 3->32->32->2 MLP, 16 samples per wave.
// K-slot permutation trick: chunk k of a 32-feature layer covers features
// {f0, f0+1, f0+8, f0+9}, f0 = base + 2*(k&3), base = (k<4)?0:16.  With that
// mapping the B-operand of every K-chunk is lane-local: lo lane s holds
// features f0,f0+1 of sample s in VGPRs 2k,2k+1 of the D-tile, hi lane s+16
// holds f0+8,f0+9 in the same VGPRs -- exactly the B layout (K0,K1 lo /
// K2,K3 hi).  No cross-lane ops anywhere in the inner loop.
__device__ __forceinline__ void ode_f(
    float tin, float y0in, float y1in, bool lo,
    const v2f A1[2], const v2f A2[2][8], const v2f A3[8],
    const v8f Cb1[2], const v8f Cb2[2], const v8f& Cb3,
    float& o0, float& o1)
{
    // ---- Layer 1: B = zin^T (K rows: t, y0, y1, 0) ----
    v2f bz; bz.x = lo ? tin : y1in; bz.y = lo ? y0in : 0.0f;
    v8f D0 = WMMA4(A1[0], bz, Cb1[0]);
    v8f D1 = WMMA4(A1[1], bz, Cb1[1]);
#pragma unroll
    for (int r = 0; r < 8; ++r) { D0[r] = fmaxf(D0[r], 0.0f); D1[r] = fmaxf(D1[r], 0.0f); }

    // ---- Layer 2: 8 lane-local K-chunks, 2 feature tiles, split accumulators ----
    v8f E0a = Cb2[0], E1a = Cb2[1];
    v8f E0b, E1b;
#pragma unroll
    for (int r = 0; r < 8; ++r) { E0b[r] = 0.0f; E1b[r] = 0.0f; }
#pragma unroll
    for (int k = 0; k < 4; ++k) {
        v2f bc; bc.x = D0[2 * k]; bc.y = D0[2 * k + 1];   // chunk k   (features from D0)
        E0a = WMMA4(A2[0][k], bc, E0a);
        E1a = WMMA4(A2[1][k], bc, E1a);
        v2f bd; bd.x = D1[2 * k]; bd.y = D1[2 * k + 1];   // chunk k+4 (features from D1)
        E0b = WMMA4(A2[0][k + 4], bd, E0b);
        E1b = WMMA4(A2[1][k + 4], bd, E1b);
    }
    v8f E0, E1;
#pragma unroll
    for (int r = 0; r < 8; ++r) {
        E0[r] = fmaxf(E0a[r] + E0b[r], 0.0f);
        E1[r] = fmaxf(E1a[r] + E1b[r], 0.0f);
    }

    // ---- Layer 3: A3 rows 8,9 duplicate rows 0,1, so the result lands in
    // VGPRs 0,1 of BOTH half-waves -> no output broadcast needed. ----
    v8f Fa = Cb3;
    v8f Fb;
#pragma unroll
    for (int r = 0; r < 8; ++r) Fb[r] = 0.0f;
#pragma unroll
    for (int k = 0; k < 4; ++k) {
        v2f bc; bc.x = E0[2 * k]; bc.y = E0[2 * k + 1];
        Fa = WMMA4(A3[k], bc, Fa);
        v2f bd; bd.x = E1[2 * k]; bd.y = E1[2 * k + 1];
        Fb = WMMA4(A3[k + 4], bd, Fb);
    }
    o0 = Fa[0] + Fb[0];
    o1 = Fa[1] + Fb[1];
}

__global__ __launch_bounds__(256) void node_rk45_kernel(
    const float* __restrict__ x, const float* __restrict__ samples,
    const float* __restrict__ w1, const float* __restrict__ b1,
    const float* __restrict__ w2, const float* __restrict__ b2,
    const float* __restrict__ w3, const float* __restrict__ b3,
    const float* __restrict__ fcw, const float* __restrict__ fcb,
    float* __restrict__ out, int B)
{
    const int lane = threadIdx.x & 31;
    const int s16  = lane & 15;
    const bool lo  = lane < 16;
    const int wave = (blockIdx.x * blockDim.x + threadIdx.x) >> 5;
    int sample = wave * 16 + s16;
    const bool valid = sample < B;
    if (sample >= B) sample = B - 1;  // clamp loads; store is guarded

    // ---- Stage weights into WMMA-layout registers (live for whole kernel) ----
    v2f A1[2], A2[2][8], A3[8];
    v8f Cb1[2], Cb2[2], Cb3;
#pragma unroll
    for (int mt = 0; mt < 2; ++mt) {
        const int j = 16 * mt + s16;              // output feature = M row
        // W1^T tiles (K: zin features t,y0 | y1,pad)
        A1[mt].x = lo ? w1[0 * 32 + j] : w1[2 * 32 + j];
        A1[mt].y = lo ? w1[1 * 32 + j] : 0.0f;
        // W2^T tiles with permuted K mapping: chunk k -> features {f0,f0+1,f0+8,f0+9}
#pragma unroll
        for (int k = 0; k < 8; ++k) {
            const int f0 = ((k < 4) ? 0 : 16) + 2 * (k & 3);
            A2[mt][k].x = lo ? w2[(f0 + 0) * 32 + j] : w2[(f0 + 8) * 32 + j];
            A2[mt][k].y = lo ? w2[(f0 + 1) * 32 + j] : w2[(f0 + 9) * 32 + j];
        }
        // Bias C-init tiles (feature-major: VGPR r = rows r / 8+r)
#pragma unroll
        for (int r = 0; r < 8; ++r) {
            Cb1[mt][r] = lo ? b1[16 * mt + r] : b1[16 * mt + 8 + r];
            Cb2[mt][r] = lo ? b2[16 * mt + r] : b2[16 * mt + 8 + r];
        }
    }
    // W3^T tiles (w3 is 32x2 row-major), same permuted K mapping.
    // Rows 0,1 = outputs; rows 8,9 duplicate them (lanes s16==8,9 stage the
    // same columns) so results appear in both half-waves.
#pragma unroll
    for (int k = 0; k < 8; ++k) {
        const int f0 = ((k < 4) ? 0 : 16) + 2 * (k & 3);
        float ax = 0.0f, ay = 0.0f;
        int jj = -1;
        if (s16 == 0 || s16 == 1) jj = s16;
        if (s16 == 8 || s16 == 9) jj = s16 - 8;
        if (jj >= 0) {
            ax = lo ? w3[(f0 + 0) * 2 + jj] : w3[(f0 + 8) * 2 + jj];
            ay = lo ? w3[(f0 + 1) * 2 + jj] : w3[(f0 + 9) * 2 + jj];
        }
        A3[k].x = ax; A3[k].y = ay;
    }
#pragma unroll
    for (int r = 0; r < 8; ++r) Cb3[r] = 0.0f;
    Cb3[0] = b3[0];   // rows 0 and 8 (duplicate) -- uniform across lanes
    Cb3[1] = b3[1];   // rows 1 and 9

    // ---- Per-sample ODE state (replicated in lane s and s+16) ----
    float y0 = x[sample * 2 + 0];
    float y1 = x[sample * 2 + 1];
    float T  = fminf(samples[sample], 50.0f);
    float t = 0.0f, hstep = 0.05f;

    // Dormand-Prince RK45 coefficients
    const float a21 = 0.2f;
    const float a31 = 3.0f / 40.0f,        a32 = 9.0f / 40.0f;
    const float a41 = 44.0f / 45.0f,       a42 = -56.0f / 15.0f,      a43 = 32.0f / 9.0f;
    const float a51 = 19372.0f / 6561.0f,  a52 = -25360.0f / 2187.0f, a53 = 64448.0f / 6561.0f, a54 = -212.0f / 729.0f;
    const float a61 = 9017.0f / 3168.0f,   a62 = -355.0f / 33.0f,     a63 = 46732.0f / 5247.0f, a64 = 49.0f / 176.0f, a65 = -5103.0f / 18656.0f;
    const float b1c = 35.0f / 384.0f,      b3c = 500.0f / 1113.0f,    b4c = 125.0f / 192.0f,    b5c = -2187.0f / 6784.0f, b6c = 11.0f / 84.0f;
    const float e1 = 71.0f / 57600.0f,     e3 = -71.0f / 16695.0f,    e4 = 71.0f / 1920.0f;
    const float e5 = -17253.0f / 339200.0f, e6 = 22.0f / 525.0f,      e7 = -1.0f / 40.0f;

    for (int step = 0; step < 96; ++step) {
        float rem   = T - t;
        bool active = rem > 1e-8f;
        float hs    = active ? fminf(hstep, rem) : 0.0f;

        float k10, k11, k20, k21, k30, k31, k40, k41, k50, k51, k60, k61, k70, k71;
        ode_f(t, y0, y1, lo, A1, A2, A3, Cb1, Cb2, Cb3, k10, k11);
        ode_f(t + hs * 0.2f,
              y0 + hs * (a21 * k10), y1 + hs * (a21 * k11),
              lo, A1, A2, A3, Cb1, Cb2, Cb3, k20, k21);
        ode_f(t + hs * 0.3f,
              y0 + hs * (a31 * k10 + a32 * k20), y1 + hs * (a31 * k11 + a32 * k21),
              lo, A1, A2, A3, Cb1, Cb2, Cb3, k30, k31);
        ode_f(t + hs * 0.8f,
              y0 + hs * (a41 * k10 + a42 * k20 + a43 * k30),
              y1 + hs * (a41 * k11 + a42 * k21 + a43 * k31),
              lo, A1, A2, A3, Cb1, Cb2, Cb3, k40, k41);
        ode_f(t + hs * (8.0f / 9.0f),
              y0 + hs * (a51 * k10 + a52 * k20 + a53 * k30 + a54 * k40),
              y1 + hs * (a51 * k11 + a52 * k21 + a53 * k31 + a54 * k41),
              lo, A1, A2, A3, Cb1, Cb2, Cb3, k50, k51);
        ode_f(t + hs,
              y0 + hs * (a61 * k10 + a62 * k20 + a63 * k30 + a64 * k40 + a65 * k50),
              y1 + hs * (a61 * k11 + a62 * k21 + a63 * k31 + a64 * k41 + a65 * k51),
              lo, A1, A2, A3, Cb1, Cb2, Cb3, k60, k61);
        float y50 = y0 + hs * (b1c * k10 + b3c * k30 + b4c * k40 + b5c * k50 + b6c * k60);
        float y51 = y1 + hs * (b1c * k11 + b3c * k31 + b4c * k41 + b5c * k51 + b6c * k61);
        ode_f(t + hs, y50, y51, lo, A1, A2, A3, Cb1, Cb2, Cb3, k70, k71);

        float err0 = hs * (e1 * k10 + e3 * k30 + e4 * k40 + e5 * k50 + e6 * k60 + e7 * k70);
        float err1 = hs * (e1 * k11 + e3 * k31 + e4 * k41 + e5 * k51 + e6 * k61 + e7 * k71);
        float sc0 = 0.01f + 0.01f * fmaxf(fabsf(y0), fabsf(y50));
        float sc1 = 0.01f + 0.01f * fmaxf(fabsf(y1), fabsf(y51));
        float r0 = err0 / sc0, r1 = err1 / sc1;
        float en = sqrtf(0.5f * (r0 * r0 + r1 * r1));
        bool accept = active && (en <= 1.0f);
        float fac = fminf(fmaxf(0.9f * powf(fmaxf(en, 1e-10f), -0.2f), 0.2f), 10.0f);
        t     = accept ? t + hs : t;
        y0    = accept ? y50 : y0;
        y1    = accept ? y51 : y1;
        hstep = active ? hs * fac : hstep;
    }

    // Classification head: act = yF @ fc_w + fc_b  (fc_w is 2x10 row-major)
    if (lo && valid) {
#pragma unroll
        for (int c = 0; c < 10; ++c) {
            out[sample * 10 + c] = fmaf(y0, fcw[c], fmaf(y1, fcw[10 + c], fcb[c]));
        }
    }
}

extern "C" void kernel_launch(void* const* d_in, const int* in_sizes, int n_in,
                              void* d_out, int out_size, void* d_ws, size_t ws_size,
                              hipStream_t stream) {
    const float* x       = (const float*)d_in[0];
    const float* samples = (const float*)d_in[1];
    const float* w1  = (const float*)d_in[2];
    const float* b1  = (const float*)d_in[3];
    const float* w2  = (const float*)d_in[4];
    const float* b2  = (const float*)d_in[5];
    const float* w3  = (const float*)d_in[6];
    const float* b3  = (const float*)d_in[7];
    const float* fcw = (const float*)d_in[8];
    const float* fcb = (const float*)d_in[9];
    // enc_* (d_in[10..15]) are dead code in the reference (alpha/beta unused).
    float* out = (float*)d_out;

    const int B = in_sizes[1];                    // samples is (B,1)
    const int samplesPerBlock = (256 / 32) * 16;  // 8 waves x 16 samples
    const int grid = (B + samplesPerBlock - 1) / samplesPerBlock;
    node_rk45_kernel<<<grid, 256, 0, stream>>>(x, samples, w1, b1, w2, b2, w3, b3,
                                               fcw, fcb, out, B);
}